// ShGaussianKernelConv_65309272703467
// MI455X (gfx1250) — compile-verified
//
#include <hip/hip_runtime.h>

typedef __attribute__((ext_vector_type(2))) float v2f;
typedef __attribute__((ext_vector_type(8))) float v8f;

// ---- problem constants (match reference setup_inputs) ----
constexpr int B_ = 2, N_ = 8192, V_ = 8192, P_ = 32, KS_ = 27; // K*S = 9*3
constexpr int CTOT = 144;           // 1*16 + 3*16 + 5*16
constexpr int BV = B_ * V_;         // 16384
constexpr int KPAD = 40;            // padded kernel-row stride in LDS (bank decorrelation)

// ---- CG table offsets (floats) in workspace ----
constexpr int OQ110 = 0,   OQ111 = 9,   OQ112 = 36,  OQ211 = 81,  OQ212 = 126,
              OQ121 = 201, OQ122 = 246, OQ220 = 321, OQ221 = 346, OQ222 = 421,
              QTOT  = 546;

// ---- output offsets in d_out (floats) ----
constexpr size_t OFF0 = 0;
constexpr size_t OFF1 = (size_t)BV * 144;
constexpr size_t OFF2 = OFF1 + (size_t)BV * 3 * 288;

// ======================================================================
// CG precompute (exact mirror of the Python host code, fp64 on device)
// ======================================================================
__device__ double dfact(int n) {
  double r = 1.0;
  for (int i = 2; i <= n; ++i) r *= (double)i;
  return r;
}
__device__ __forceinline__ int imax2(int a, int b) { return a > b ? a : b; }
__device__ __forceinline__ int imin2(int a, int b) { return a < b ? a : b; }
__device__ __forceinline__ int iabs_(int a) { return a < 0 ? -a : a; }

__device__ double cg_cplx(int j1, int m1, int j2, int m2, int j3, int m3) {
  if (m3 != m1 + m2) return 0.0;
  if (j3 < iabs_(j1 - j2) || j3 > j1 + j2) return 0.0;
  double pre = sqrt((2.0 * j3 + 1.0) * dfact(j1 + j2 - j3) * dfact(j1 - j2 + j3) *
                    dfact(-j1 + j2 + j3) / dfact(j1 + j2 + j3 + 1));
  pre *= sqrt(dfact(j3 + m3) * dfact(j3 - m3) * dfact(j1 - m1) * dfact(j1 + m1) *
              dfact(j2 - m2) * dfact(j2 + m2));
  int kmin = imax2(0, imax2(j2 - j3 - m1, j1 - j3 + m2));
  int kmax = imin2(j1 + j2 - j3, imin2(j1 - m1, j2 + m2));
  double s = 0.0;
  for (int k = kmin; k <= kmax; ++k) {
    double d = dfact(k) * dfact(j1 + j2 - j3 - k) * dfact(j1 - m1 - k) *
               dfact(j2 + m2 - k) * dfact(j3 - j2 + m1 + k) * dfact(j3 - j1 - m2 + k);
    s += ((k & 1) ? -1.0 : 1.0) / d;
  }
  return pre * s;
}

// real spherical harmonic change of basis, (2l+1)x(2l+1) complex
__device__ void real_basis(int l, double* Ar, double* Ai) {
  const int d = 2 * l + 1;
  for (int i = 0; i < d * d; ++i) { Ar[i] = 0.0; Ai[i] = 0.0; }
  const double s2 = 1.0 / sqrt(2.0);
  for (int m = -l; m <= l; ++m) {
    const double sg = (m & 1) ? -1.0 : 1.0;   // (-1)^m
    if (m < 0) {
      Ai[(l + m) * d + (l + m)] = s2;
      Ai[(l + m) * d + (l - m)] = -sg * s2;
    } else if (m == 0) {
      Ar[l * d + l] = 1.0;
    } else {
      Ar[(l + m) * d + (l - m)] = s2;
      Ar[(l + m) * d + (l + m)] = sg * s2;
    }
  }
}

__device__ void build_Q(int j, int l, int J, float* out) {
  double Ajr[25], Aji[25], Alr[25], Ali[25], AJr[25], AJi[25];
  real_basis(j, Ajr, Aji);
  real_basis(l, Alr, Ali);
  real_basis(J, AJr, AJi);
  const int dj = 2 * j + 1, dl = 2 * l + 1, dJ = 2 * J + 1;
  double Qr[125], Qi[125];
  double maxR = 0.0, maxI = 0.0;
  for (int M = 0; M < dJ; ++M)
    for (int a = 0; a < dj; ++a)
      for (int b = 0; b < dl; ++b) {
        double sr = 0.0, si = 0.0;
        for (int p = 0; p < dj; ++p) {
          const double ar = Ajr[a * dj + p], ai = -Aji[a * dj + p];   // conj
          if (ar == 0.0 && ai == 0.0) continue;
          for (int q = 0; q < dl; ++q) {
            const double br = Alr[b * dl + q], bi = -Ali[b * dl + q]; // conj
            if (br == 0.0 && bi == 0.0) continue;
            for (int Nn = 0; Nn < dJ; ++Nn) {
              const double c = cg_cplx(j, p - j, l, q - l, J, Nn - J);
              if (c == 0.0) continue;
              const double jr = AJr[M * dJ + Nn], ji = AJi[M * dJ + Nn];
              if (jr == 0.0 && ji == 0.0) continue;
              const double t1r = jr * ar - ji * ai, t1i = jr * ai + ji * ar;
              const double tr = t1r * br - t1i * bi, ti = t1r * bi + t1i * br;
              sr += tr * c;
              si += ti * c;
            }
          }
        }
        Qr[(M * dj + a) * dl + b] = sr;
        Qi[(M * dj + a) * dl + b] = si;
        if (fabs(sr) > maxR) maxR = fabs(sr);
        if (fabs(si) > maxI) maxI = fabs(si);
      }
  const int n = dJ * dj * dl;
  if (maxR >= maxI)
    for (int i = 0; i < n; ++i) out[i] = (float)Qr[i];
  else
    for (int i = 0; i < n; ++i) out[i] = (float)Qi[i];
}

__global__ void cg_init_kernel(float* __restrict__ ws) {
  if (threadIdx.x != 0 || blockIdx.x != 0) return;
  build_Q(1, 1, 0, ws + OQ110);
  build_Q(1, 1, 1, ws + OQ111);
  build_Q(1, 1, 2, ws + OQ112);
  build_Q(2, 1, 1, ws + OQ211);
  build_Q(2, 1, 2, ws + OQ212);
  build_Q(1, 2, 1, ws + OQ121);
  build_Q(1, 2, 2, ws + OQ122);
  build_Q(2, 2, 0, ws + OQ220);
  build_Q(2, 2, 1, ws + OQ221);
  build_Q(2, 2, 2, ws + OQ222);
}

// ======================================================================
// Main kernel: one wave32 per (b, v)
// ======================================================================
__device__ __forceinline__
void cg_apply(const float* __restrict__ Q, int rowsJ, int dj, int dl,
              int kbase, int colbase, const float* __restrict__ yS,
              float* __restrict__ outp, int ostride, int ocol, int lane)
{
  const int total = rowsJ * 48;
  for (int e = lane; e < total; e += 32) {
    const int M = e / 48, c = e % 48;
    const int s = c >> 4, ch = c & 15;
    float sum = 0.f;
    for (int p = 0; p < dj; ++p)
      for (int q = 0; q < dl; ++q)
        sum += Q[(M * dj + p) * dl + q] *
               yS[((kbase + p) * 3 + s) * CTOT + colbase + q * 16 + ch];
    outp[M * ostride + ocol + c] = sum;
  }
}

__global__ __launch_bounds__(32)
void sh_conv_kernel(const float* __restrict__ feat0,
                    const float* __restrict__ feat1,
                    const float* __restrict__ feat2,
                    const float* __restrict__ kern,
                    const int*   __restrict__ pidx,
                    const float* __restrict__ qtab,
                    float* __restrict__ out)
{
  __shared__ __align__(16) float sigS[P_ * CTOT];   // gathered signal, 32x144
  __shared__ float kernS[P_ * KPAD];                // kernel rows, 32x40 (cols 27..31 zero)
  __shared__ float yS[KS_ * CTOT];                  // y = kern^T * sig, 27x144
  __shared__ float qS[QTOT];                        // CG tables

  const int lane = threadIdx.x;
  const int lo = lane & 15, hi = lane >> 4;
  const int bv = blockIdx.x;
  const int b  = bv >> 13;                          // V = 8192

  // stage CG tables into LDS
  for (int i = lane; i < QTOT; i += 32) qS[i] = qtab[i];

  // ---- gather: lane owns patch element p = lane ----
  {
    const int n = pidx[(size_t)bv * P_ + lane];
    const size_t row = (size_t)b * N_ + (size_t)n;
    const float4* s0 = (const float4*)(feat0 + row * 16);
    const float4* s1 = (const float4*)(feat1 + row * 48);
    const float4* s2 = (const float4*)(feat2 + row * 80);
    float4* dst = (float4*)(sigS + lane * CTOT);
#pragma unroll
    for (int i = 0; i < 4; ++i)  dst[i]      = s0[i];
#pragma unroll
    for (int i = 0; i < 12; ++i) dst[4 + i]  = s1[i];
#pragma unroll
    for (int i = 0; i < 20; ++i) dst[16 + i] = s2[i];

    const float* kr = kern + ((size_t)bv * P_ + lane) * KS_;
#pragma unroll
    for (int y = 0; y < KS_; ++y) kernS[lane * KPAD + y] = kr[y];
#pragma unroll
    for (int y = KS_; y < 32; ++y) kernS[lane * KPAD + y] = 0.f;
  }
  __syncthreads();

  // ---- y(27x144) = kern^T(27x32) * sig(32x144) via v_wmma_f32_16x16x4_f32 ----
  // A 16x4 fp32 layout: lanes 0-15 -> M=lane, K={0,1}; lanes 16-31 -> M=lane-16, K={2,3}
  // C/D layout: VGPR r -> M = 16*mt + 8*hi + r, N = 16*nt + lo

  // Preload all A fragments once (8 K-steps x 2 M-tiles = 32 VGPRs)
  v2f afr[8][2];
#pragma unroll
  for (int kk = 0; kk < 8; ++kk) {
    const int p0 = kk * 4 + hi * 2;
    afr[kk][0].x = kernS[p0 * KPAD + lo];
    afr[kk][0].y = kernS[(p0 + 1) * KPAD + lo];
    afr[kk][1].x = kernS[p0 * KPAD + 16 + lo];
    afr[kk][1].y = kernS[(p0 + 1) * KPAD + 16 + lo];
  }

  // Outer loop over N-tiles keeps only 2 accumulators live at a time
#pragma unroll
  for (int nt = 0; nt < 9; ++nt) {
    v8f acc0 = {};
    v8f acc1 = {};
#pragma unroll
    for (int kk = 0; kk < 8; ++kk) {
      const int p0 = kk * 4 + hi * 2;
      v2f bfr;
      bfr.x = sigS[p0 * CTOT + nt * 16 + lo];
      bfr.y = sigS[(p0 + 1) * CTOT + nt * 16 + lo];
      acc0 = __builtin_amdgcn_wmma_f32_16x16x4_f32(
          false, afr[kk][0], false, bfr, (short)0, acc0, false, false);
      acc1 = __builtin_amdgcn_wmma_f32_16x16x4_f32(
          false, afr[kk][1], false, bfr, (short)0, acc1, false, false);
    }
    // spill finished tiles to LDS (drop M>=27 padding rows of the second tile)
#pragma unroll
    for (int r = 0; r < 8; ++r) {
      const int M0 = hi * 8 + r;                 // 0..15, always valid
      yS[M0 * CTOT + nt * 16 + lo] = acc0[r];
      const int M1 = 16 + hi * 8 + r;            // 16..31
      if (M1 < KS_)
        yS[M1 * CTOT + nt * 16 + lo] = acc1[r];
    }
  }
  __syncthreads();

  // ---- CG stage: y rows indexed (k*3 + s), cols (off_l + m*16 + ch) ----
  float* o0 = out + OFF0 + (size_t)bv * 144;
  float* o1 = out + OFF1 + (size_t)bv * 3 * 288;
  float* o2 = out + OFF2 + (size_t)bv * 5 * 288;

  // out0: [ (l=0,j=0) | (1x1->J=0) | (2x2->J=0) ]
  for (int e = lane; e < 48; e += 32) {
    const int s = e >> 4, ch = e & 15;
    o0[e] = yS[s * CTOT + ch];
  }
  cg_apply(qS + OQ110, 1, 3, 3, /*k0=*/1, /*col0=*/16, yS, o0, 144, 48, lane);
  cg_apply(qS + OQ220, 1, 5, 5, 4, 64, yS, o0, 144, 96, lane);

  // out1: [ (l=0,j=1) | (l=1,j=0) | 1x1->1 | 2x1->1 | 1x2->1 | 2x2->1 ]
  for (int e = lane; e < 3 * 48; e += 32) {
    const int M = e / 48, c = e % 48, s = c >> 4, ch = c & 15;
    o1[M * 288 + c]      = yS[((1 + M) * 3 + s) * CTOT + ch];     // j=1, l=0
    o1[M * 288 + 48 + c] = yS[s * CTOT + 16 + M * 16 + ch];       // j=0, l=1
  }
  cg_apply(qS + OQ111, 3, 3, 3, 1, 16, yS, o1, 288, 96,  lane);
  cg_apply(qS + OQ211, 3, 5, 3, 4, 16, yS, o1, 288, 144, lane);
  cg_apply(qS + OQ121, 3, 3, 5, 1, 64, yS, o1, 288, 192, lane);
  cg_apply(qS + OQ221, 3, 5, 5, 4, 64, yS, o1, 288, 240, lane);

  // out2: [ (l=0,j=2) | (l=2,j=0) | 1x1->2 | 2x1->2 | 1x2->2 | 2x2->2 ]
  for (int e = lane; e < 5 * 48; e += 32) {
    const int M = e / 48, c = e % 48, s = c >> 4, ch = c & 15;
    o2[M * 288 + c]      = yS[((4 + M) * 3 + s) * CTOT + ch];     // j=2, l=0
    o2[M * 288 + 48 + c] = yS[s * CTOT + 64 + M * 16 + ch];       // j=0, l=2
  }
  cg_apply(qS + OQ112, 5, 3, 3, 1, 16, yS, o2, 288, 96,  lane);
  cg_apply(qS + OQ212, 5, 5, 3, 4, 16, yS, o2, 288, 144, lane);
  cg_apply(qS + OQ122, 5, 3, 5, 1, 64, yS, o2, 288, 192, lane);
  cg_apply(qS + OQ222, 5, 5, 5, 4, 64, yS, o2, 288, 240, lane);
}

// ======================================================================
extern "C" void kernel_launch(void* const* d_in, const int* in_sizes, int n_in,
                              void* d_out, int out_size, void* d_ws, size_t ws_size,
                              hipStream_t stream)
{
  (void)in_sizes; (void)n_in; (void)out_size; (void)ws_size;
  const float* feat0 = (const float*)d_in[0];
  const float* feat1 = (const float*)d_in[1];
  const float* feat2 = (const float*)d_in[2];
  const float* kern  = (const float*)d_in[3];
  const int*   pidx  = (const int*)d_in[4];
  float* qtab = (float*)d_ws;   // 546 floats of CG tables

  cg_init_kernel<<<1, 1, 0, stream>>>(qtab);
  sh_conv_kernel<<<dim3(BV), dim3(32), 0, stream>>>(
      feat0, feat1, feat2, kern, pidx, qtab, (float*)d_out);
}